// myDC_79405355368487
// MI455X (gfx1250) — compile-verified
//
#include <hip/hip_runtime.h>

// ---------------------------------------------------------------------------
// CG-SENSE on MI455X (gfx1250): all DFTs expressed as complex GEMMs on the
// fp32 WMMA pipe (V_WMMA_F32_16X16X4_F32). Transform lengths 288/96/48 are
// multiples of 48, so each wave computes a 48x16 output tile (3 m-tiles,
// 12 WMMAs per k-step). Strides are template constants so all X-tile loads
// use immediate offsets (no per-load 64-bit VALU address adds).
// ---------------------------------------------------------------------------

typedef float v2f __attribute__((ext_vector_type(2)));
typedef float v8f __attribute__((ext_vector_type(8)));

#define NRI   96
#define RP    288
#define NCOLS 96
#define NSL   48
#define XDIM  (NCOLS*NSL)     /* 4608  */
#define NV    (NRI*XDIM)      /* 442368: image-domain vector length */
#define NBB   (RP*XDIM)       /* 1327104: padded coil buffer length */
#define NCOIL 12
#define NBATCH 2
#define CG_ITERS 10
#define CG_TOL 1e-5f

// ---------------------------------------------------------------------------
// DFT matrix generation: W[m,k] = scale * exp(sgn * 2*pi*i * m*k / N)
// fwd: sgn=-1, scale=1 ; inv: sgn=+1, scale=1/N   (matches jnp.fft)
// ---------------------------------------------------------------------------
__global__ void init_dft_kernel(float* __restrict__ Wr, float* __restrict__ Wi,
                                int N, float sgn, float scale) {
    int idx = blockIdx.x * blockDim.x + threadIdx.x;
    if (idx >= N * N) return;
    int m = idx / N, k = idx % N;
    double ang = (double)sgn * 6.283185307179586 * (double)((m * k) % N) / (double)N;
    Wr[idx] = (float)cos(ang) * scale;
    Wi[idx] = (float)sin(ang) * scale;
}

// ---------------------------------------------------------------------------
// Complex DFT-as-GEMM tile kernel. One wave per 48x16 output tile (MT=3
// m-tiles). Y[m,n] = sum_k W[m,k] * X[k,n] (complex). X/Y addressed as
//   base + k*SK + n*SN  (elements), base = blockIdx.z * outerStride.
// Fragments follow ISA 7.12.2: f32 A 16x4 -> 2 VGPRs, lane halves hold
// K={0,1} / K={2,3}; C/D: VGPR i holds M = i (+8 for upper lane half).
// SK/SN are compile-time so X loads fold k*SK into immediate offsets.
// ---------------------------------------------------------------------------
template<int NLEN, int SK, int SN, int MT>
__global__ __launch_bounds__(32)
void dft_wmma_kernel(const float* __restrict__ Wr, const float* __restrict__ Wi,
                     const float* __restrict__ Xr, const float* __restrict__ Xi,
                     float* __restrict__ Yr, float* __restrict__ Yi,
                     long outerStride) {
    const int lane = threadIdx.x;
    const int half = lane >> 4;
    const int l15  = lane & 15;
    const int m0 = blockIdx.y * (16 * MT);
    const int n0 = blockIdx.x * 16;
    const long base = (long)blockIdx.z * outerStride;

    v8f acc_rr[MT], acc_ii[MT], acc_ri[MT], acc_ir[MT];
#pragma unroll
    for (int t = 0; t < MT; ++t) {
        acc_rr[t] = (v8f){}; acc_ii[t] = (v8f){};
        acc_ri[t] = (v8f){}; acc_ir[t] = (v8f){};
    }

    const int ak = half * 2;                       // K sub-offset per lane half
    const float* wr_row[MT];
    const float* wi_row[MT];
#pragma unroll
    for (int t = 0; t < MT; ++t) {
        wr_row[t] = Wr + (long)(m0 + 16 * t + l15) * NLEN;
        wi_row[t] = Wi + (long)(m0 + 16 * t + l15) * NLEN;
    }
    const float* xr_p = Xr + base + (long)(n0 + l15) * SN + (long)ak * SK;
    const float* xi_p = Xi + base + (long)(n0 + l15) * SN + (long)ak * SK;

#pragma unroll 4
    for (int k = 0; k < NLEN; k += 4) {
        v2f br, bi;
        br.x = xr_p[(long)k * SK];       br.y = xr_p[(long)k * SK + SK];
        bi.x = xi_p[(long)k * SK];       bi.y = xi_p[(long)k * SK + SK];
#pragma unroll
        for (int t = 0; t < MT; ++t) {
            v2f ar, ai;
            ar.x = wr_row[t][k + ak];  ar.y = wr_row[t][k + ak + 1];
            ai.x = wi_row[t][k + ak];  ai.y = wi_row[t][k + ak + 1];
            acc_rr[t] = __builtin_amdgcn_wmma_f32_16x16x4_f32(false, ar, false, br, (short)0, acc_rr[t], false, false);
            acc_ii[t] = __builtin_amdgcn_wmma_f32_16x16x4_f32(false, ai, false, bi, (short)0, acc_ii[t], false, false);
            acc_ri[t] = __builtin_amdgcn_wmma_f32_16x16x4_f32(false, ar, false, bi, (short)0, acc_ri[t], false, false);
            acc_ir[t] = __builtin_amdgcn_wmma_f32_16x16x4_f32(false, ai, false, br, (short)0, acc_ir[t], false, false);
        }
    }

    const int mh = half * 8;
#pragma unroll
    for (int t = 0; t < MT; ++t) {
#pragma unroll
        for (int i = 0; i < 8; ++i) {
            long off = base + (long)(m0 + 16 * t + mh + i) * SK + (long)(n0 + l15) * SN;
            Yr[off] = acc_rr[t][i] - acc_ii[t][i];   // Re = Wr*Xr - Wi*Xi
            Yi[off] = acc_ri[t][i] + acc_ir[t][i];   // Im = Wr*Xi + Wi*Xr
        }
    }
}

// ---------------------------------------------------------------------------
// Pointwise kernels
// ---------------------------------------------------------------------------
__global__ void pack_rhs_kernel(const float* __restrict__ rhs,
                                float* rr, float* ri, float* pr, float* pi,
                                float* xr, float* xi, long base2, int n) {
    int i = blockIdx.x * blockDim.x + threadIdx.x;
    if (i >= n) return;
    float re = rhs[base2 + 2L * i], im = rhs[base2 + 2L * i + 1];
    rr[i] = re; ri[i] = im; pr[i] = re; pi[i] = im; xr[i] = 0.f; xi[i] = 0.f;
}

// q = pad(csm * p): rows [96,192) hold the image, rest zero
__global__ void expand_pad_kernel(const float* __restrict__ csm,
                                  const float* __restrict__ pr, const float* __restrict__ pi,
                                  float* __restrict__ br, float* __restrict__ bi, long csm_base) {
    int i = blockIdx.x * blockDim.x + threadIdx.x;
    if (i >= NBB) return;
    int row = i / XDIM;
    if (row < NRI || row >= 2 * NRI) { br[i] = 0.f; bi[i] = 0.f; return; }
    int j = (row - NRI) * XDIM + (i % XDIM);
    float cr = csm[csm_base + 2L * j], ci = csm[csm_base + 2L * j + 1];
    float qr = pr[j], qi = pi[j];
    br[i] = cr * qr - ci * qi;
    bi[i] = cr * qi + ci * qr;
}

__global__ void mul_wpsf_kernel(float* __restrict__ br, float* __restrict__ bi,
                                const float* __restrict__ w, long wbase, float conjsgn) {
    int i = blockIdx.x * blockDim.x + threadIdx.x;
    if (i >= NBB) return;
    float wr = w[wbase + 2L * i], wi = conjsgn * w[wbase + 2L * i + 1];
    float ur = br[i], ui = bi[i];
    br[i] = wr * ur - wi * ui;
    bi[i] = wr * ui + wi * ur;
}

__global__ void mul_mask_kernel(float* __restrict__ br, float* __restrict__ bi,
                                const float* __restrict__ mask, long mbase) {
    int i = blockIdx.x * blockDim.x + threadIdx.x;
    if (i >= NBB) return;
    float m = mask[mbase + i];
    br[i] *= m; bi[i] *= m;
}

__global__ void init_ap_kernel(float* apr, float* api,
                               const float* pr, const float* pi,
                               const float* __restrict__ lam, int n) {
    int i = blockIdx.x * blockDim.x + threadIdx.x;
    if (i >= n) return;
    float l = lam[0];
    apr[i] = l * pr[i];
    api[i] = l * pi[i];
}

// Ap += crop(v) * conj(csm)
__global__ void combine_acc_kernel(float* __restrict__ apr, float* __restrict__ api,
                                   const float* __restrict__ vr, const float* __restrict__ vi,
                                   const float* __restrict__ csm, long csm_base) {
    int i = blockIdx.x * blockDim.x + threadIdx.x;
    if (i >= NV) return;
    long src = (long)i + (long)NRI * XDIM;    // crop rows [96,192)
    float cr = csm[csm_base + 2L * i], ci = csm[csm_base + 2L * i + 1];
    float ur = vr[src], ui = vi[src];
    apr[i] += ur * cr + ui * ci;              // u * conj(c)
    api[i] += ui * cr - ur * ci;
}

// ---------------------------------------------------------------------------
// Deterministic two-stage reduction: sum(a*b + c*d)
// ---------------------------------------------------------------------------
__global__ void dot_stage1_kernel(const float* a, const float* b,
                                  const float* c, const float* d,
                                  float* __restrict__ partial, int n) {
    __shared__ float s[256];
    float acc = 0.f;
    for (int i = blockIdx.x * blockDim.x + threadIdx.x; i < n; i += gridDim.x * blockDim.x)
        acc += a[i] * b[i] + c[i] * d[i];
    s[threadIdx.x] = acc;
    __syncthreads();
    for (int off = 128; off > 0; off >>= 1) {
        if ((int)threadIdx.x < off) s[threadIdx.x] += s[threadIdx.x + off];
        __syncthreads();
    }
    if (threadIdx.x == 0) partial[blockIdx.x] = s[0];
}

__global__ void dot_stage2_kernel(const float* __restrict__ partial, float* outv) {
    __shared__ float s[256];
    s[threadIdx.x] = partial[threadIdx.x];
    __syncthreads();
    for (int off = 128; off > 0; off >>= 1) {
        if ((int)threadIdx.x < off) s[threadIdx.x] += s[threadIdx.x + off];
        __syncthreads();
    }
    if (threadIdx.x == 0) outv[0] = s[0];
}

// ---------------------------------------------------------------------------
// CG scalar control (device-side, so the full loop graph-captures)
// S[0]=rTr  S[1]=pAp  S[2]=alpha_eff  S[3]=beta  S[4]=act  S[5]=rTr_new
// ---------------------------------------------------------------------------
__global__ void cg_alpha_kernel(float* S) {
    float act = (S[0] > CG_TOL) ? 1.f : 0.f;
    S[4] = act;
    S[2] = (act != 0.f) ? (S[0] / S[1]) : 0.f;
}

__global__ void cg_beta_kernel(float* S) {
    S[3] = S[5] / S[0];
    if (S[4] != 0.f) S[0] = S[5];
}

__global__ void cg_axpy_kernel(float* xr, float* xi, float* rr, float* ri,
                               const float* pr, const float* pi,
                               const float* apr, const float* api,
                               const float* __restrict__ S, int n) {
    int i = blockIdx.x * blockDim.x + threadIdx.x;
    if (i >= n) return;
    float a = S[2];
    xr[i] += a * pr[i];   xi[i] += a * pi[i];
    rr[i] -= a * apr[i];  ri[i] -= a * api[i];
}

__global__ void cg_update_p_kernel(float* pr, float* pi,
                                   const float* rr, const float* ri,
                                   const float* __restrict__ S, int n) {
    int i = blockIdx.x * blockDim.x + threadIdx.x;
    if (i >= n) return;
    if (S[4] != 0.f) {
        float b = S[3];
        pr[i] = rr[i] + b * pr[i];
        pi[i] = ri[i] + b * pi[i];
    }
}

__global__ void write_out_kernel(float* __restrict__ out,
                                 const float* xr, const float* xi, long base2, int n) {
    int i = blockIdx.x * blockDim.x + threadIdx.x;
    if (i >= n) return;
    out[base2 + 2L * i]     = xr[i];
    out[base2 + 2L * i + 1] = xi[i];
}

// ---------------------------------------------------------------------------
extern "C" void kernel_launch(void* const* d_in, const int* in_sizes, int n_in,
                              void* d_out, int out_size, void* d_ws, size_t ws_size,
                              hipStream_t stream) {
    (void)in_sizes; (void)n_in; (void)out_size; (void)ws_size;
    const float* rhs  = (const float*)d_in[0];
    const float* csm  = (const float*)d_in[1];
    const float* mask = (const float*)d_in[2];
    const float* wpsf = (const float*)d_in[3];
    const float* lam  = (const float*)d_in[4];
    float* out = (float*)d_out;

    // ---- workspace carve-up (floats) ----
    float* w = (float*)d_ws;
    size_t off = 0;
    auto alloc = [&](size_t n) { float* p = w + off; off += n; return p; };
    float* W288fr = alloc(RP * RP);     float* W288fi = alloc(RP * RP);
    float* W288ir = alloc(RP * RP);     float* W288ii = alloc(RP * RP);
    float* W96fr  = alloc(NCOLS*NCOLS); float* W96fi  = alloc(NCOLS*NCOLS);
    float* W96ir  = alloc(NCOLS*NCOLS); float* W96ii  = alloc(NCOLS*NCOLS);
    float* W48fr  = alloc(NSL*NSL);     float* W48fi  = alloc(NSL*NSL);
    float* W48ir  = alloc(NSL*NSL);     float* W48ii  = alloc(NSL*NSL);
    float* b1r = alloc(NBB);  float* b1i = alloc(NBB);   // ping-pong coil buffers
    float* b2r = alloc(NBB);  float* b2i = alloc(NBB);
    float* xr_ = alloc(NV);   float* xi_ = alloc(NV);
    float* rr_ = alloc(NV);   float* ri_ = alloc(NV);
    float* pr_ = alloc(NV);   float* pi_ = alloc(NV);
    float* apr_ = alloc(NV);  float* api_ = alloc(NV);
    float* partial = alloc(256);
    float* S = alloc(16);

    const int blocksNB = (NBB + 255) / 256;
    const int blocksNV = (NV + 255) / 256;

    // ---- DFT matrices ----
    init_dft_kernel<<<(RP*RP + 255)/256, 256, 0, stream>>>(W288fr, W288fi, RP, -1.f, 1.f);
    init_dft_kernel<<<(RP*RP + 255)/256, 256, 0, stream>>>(W288ir, W288ii, RP,  1.f, 1.f / RP);
    init_dft_kernel<<<(NCOLS*NCOLS + 255)/256, 256, 0, stream>>>(W96fr, W96fi, NCOLS, -1.f, 1.f);
    init_dft_kernel<<<(NCOLS*NCOLS + 255)/256, 256, 0, stream>>>(W96ir, W96ii, NCOLS,  1.f, 1.f / NCOLS);
    init_dft_kernel<<<(NSL*NSL + 255)/256, 256, 0, stream>>>(W48fr, W48fi, NSL, -1.f, 1.f);
    init_dft_kernel<<<(NSL*NSL + 255)/256, 256, 0, stream>>>(W48ir, W48ii, NSL,  1.f, 1.f / NSL);

    // DFT launch geometry: each wave covers 48 M-rows (MT=3)
    const dim3 gRow(XDIM / 16, RP / 48, 1);        // transform over 288 rows
    const dim3 gCol(NSL / 16, NCOLS / 48, RP);     // transform over 96 cols per row-plane
    const dim3 gSlc(NCOLS / 16, NSL / 48, RP);     // transform over 48 slices per row-plane

    for (int b = 0; b < NBATCH; ++b) {
        const long rhsBase  = (long)b * NV * 2;
        const long wpsfBase = (long)b * NBB * 2;
        const long maskBase = (long)b * NBB;

        pack_rhs_kernel<<<blocksNV, 256, 0, stream>>>(rhs, rr_, ri_, pr_, pi_, xr_, xi_, rhsBase, NV);
        dot_stage1_kernel<<<256, 256, 0, stream>>>(rr_, rr_, ri_, ri_, partial, NV);
        dot_stage2_kernel<<<1, 256, 0, stream>>>(partial, S + 0);   // rTr0

        for (int it = 0; it < CG_ITERS; ++it) {
            init_ap_kernel<<<blocksNV, 256, 0, stream>>>(apr_, api_, pr_, pi_, lam, NV);

            for (int c = 0; c < NCOIL; ++c) {
                const long csmBase = (long)(b * NCOIL + c) * NV * 2;
                expand_pad_kernel<<<blocksNB, 256, 0, stream>>>(csm, pr_, pi_, b1r, b1i, csmBase);
                // forward row DFT (288): b1 -> b2
                dft_wmma_kernel<RP, XDIM, 1, 3><<<gRow, 32, 0, stream>>>(
                    W288fr, W288fi, b1r, b1i, b2r, b2i, 0L);
                mul_wpsf_kernel<<<blocksNB, 256, 0, stream>>>(b2r, b2i, wpsf, wpsfBase, 1.f);
                // forward col DFT (96): b2 -> b1
                dft_wmma_kernel<NCOLS, NSL, 1, 3><<<gCol, 32, 0, stream>>>(
                    W96fr, W96fi, b2r, b2i, b1r, b1i, (long)XDIM);
                // forward slice DFT (48): b1 -> b2
                dft_wmma_kernel<NSL, 1, NSL, 3><<<gSlc, 32, 0, stream>>>(
                    W48fr, W48fi, b1r, b1i, b2r, b2i, (long)XDIM);
                mul_mask_kernel<<<blocksNB, 256, 0, stream>>>(b2r, b2i, mask, maskBase);
                // inverse slice DFT: b2 -> b1
                dft_wmma_kernel<NSL, 1, NSL, 3><<<gSlc, 32, 0, stream>>>(
                    W48ir, W48ii, b2r, b2i, b1r, b1i, (long)XDIM);
                // inverse col DFT: b1 -> b2
                dft_wmma_kernel<NCOLS, NSL, 1, 3><<<gCol, 32, 0, stream>>>(
                    W96ir, W96ii, b1r, b1i, b2r, b2i, (long)XDIM);
                mul_wpsf_kernel<<<blocksNB, 256, 0, stream>>>(b2r, b2i, wpsf, wpsfBase, -1.f);
                // inverse row DFT (288): b2 -> b1
                dft_wmma_kernel<RP, XDIM, 1, 3><<<gRow, 32, 0, stream>>>(
                    W288ir, W288ii, b2r, b2i, b1r, b1i, 0L);
                combine_acc_kernel<<<blocksNV, 256, 0, stream>>>(apr_, api_, b1r, b1i, csm, csmBase);
            }

            // alpha = rTr / Re<p, Ap>
            dot_stage1_kernel<<<256, 256, 0, stream>>>(pr_, apr_, pi_, api_, partial, NV);
            dot_stage2_kernel<<<1, 256, 0, stream>>>(partial, S + 1);
            cg_alpha_kernel<<<1, 1, 0, stream>>>(S);
            cg_axpy_kernel<<<blocksNV, 256, 0, stream>>>(xr_, xi_, rr_, ri_, pr_, pi_, apr_, api_, S, NV);
            // rTr_new
            dot_stage1_kernel<<<256, 256, 0, stream>>>(rr_, rr_, ri_, ri_, partial, NV);
            dot_stage2_kernel<<<1, 256, 0, stream>>>(partial, S + 5);
            cg_beta_kernel<<<1, 1, 0, stream>>>(S);
            cg_update_p_kernel<<<blocksNV, 256, 0, stream>>>(pr_, pi_, rr_, ri_, S, NV);
        }

        write_out_kernel<<<blocksNV, 256, 0, stream>>>(out, xr_, xi_, rhsBase, NV);
    }
}